// ForceStressOutputFromEdge_18382460027058
// MI455X (gfx1250) — compile-verified
//
#include <hip/hip_runtime.h>

// ---------------------------------------------------------------------------
// ForceStressOutputFromEdge for MI455X (gfx1250)
//
// Memory-bound scatter-add: ~230 MB streamed @ 23.3 TB/s ~= 10 us floor.
// CDNA5 paths used:
//   * global_load_async_to_lds_b128 + s_wait_asynccnt : double-buffered
//     per-wave streaming of rij/fij tiles into LDS (ASYNCcnt pipeline).
//   * ds_add_f32 LDS atomics into PER-WAVE 32x6 stress bins (no cross-wave
//     LDS atomic contention), block-reduced once at the end.
//   * global_atomic_add_f32 for the force scatter (L2-resident, 1.2 MB).
// ---------------------------------------------------------------------------

#define K_NODES 100000
#define K_EDGES 6400000
#define K_BATCH 32
#define NBINS   (K_BATCH * 6)   // 192

#define WPB 8            // waves per block (wave32)
#define BT  256          // threads per block
#define TE  128          // edges per wave-tile
#define TF  (TE * 3)     // floats per array per tile (384)
#define TB  (TF * 4)     // bytes  per array per tile (1536 = 3 x 32 lanes x 16B)
#define NB  1024         // blocks

// Low 32 bits of a generic pointer to __shared__ == workgroup-relative LDS
// byte address (aperture lives in the high 32 bits), which is exactly what
// the VDST operand of global_load_async_to_lds expects.
__device__ __forceinline__ unsigned lds_off(const void* p) {
  return (unsigned)(unsigned long long)p;
}

__device__ __forceinline__ void async_b128(unsigned lds, const void* g) {
  unsigned long long ga = (unsigned long long)g;
  asm volatile("global_load_async_to_lds_b128 %0, %1, off"
               :: "v"(lds), "v"(ga) : "memory");
}

// 6 async B128 loads per tile: 3 cover rij (1536B), 3 cover fij (1536B).
__device__ __forceinline__ void prefetch_tile(const char* rijB, const char* fijB,
                                              int tile, unsigned ldsR, unsigned ldsF,
                                              int lane) {
  long long base = (long long)tile * TB + (long long)lane * 16;
#pragma unroll
  for (int j = 0; j < 3; ++j) {
    async_b128(ldsR + (unsigned)(j * 512 + lane * 16), rijB + base + j * 512);
    async_b128(ldsF + (unsigned)(j * 512 + lane * 16), fijB + base + j * 512);
  }
}

__device__ __forceinline__ void fatomic(float* p, float v) {
  unsafeAtomicAdd(p, v);   // hardware f32 atomic add (no CAS loop)
}

__global__ __launch_bounds__(BT) void edge_scatter_kernel(
    const float* __restrict__ rij, const float* __restrict__ fij,
    const int* __restrict__ esrc, const int* __restrict__ edst,
    const int* __restrict__ batch,
    float* __restrict__ force,          // [N_NODES*3], pre-zeroed
    float* __restrict__ acc) {          // [32*6] global stress accumulator, pre-zeroed
  __shared__ float smem[WPB * 2 * (2 * TF)];   // per-wave double buffer: rij|fij
  __shared__ float sAcc[WPB][NBINS];           // per-wave stress bins

  const int tid  = threadIdx.x;
  const int lane = tid & 31;
  const int w    = tid >> 5;
  const int waveId     = blockIdx.x * WPB + w;
  const int totalWaves = gridDim.x * WPB;
  const int nTiles     = K_EDGES / TE;         // 50000, exact

  // Each wave zeroes its private bins (wave-private: no barrier needed yet).
  for (int i = lane; i < NBINS; i += 32) sAcc[w][i] = 0.0f;
  float* myBins = &sAcc[w][0];

  float* buf0 = &smem[w * 2 * (2 * TF)];
  float* buf1 = buf0 + 2 * TF;
  const unsigned ldsR0 = lds_off(buf0), ldsF0 = lds_off(buf0 + TF);
  const unsigned ldsR1 = lds_off(buf1), ldsF1 = lds_off(buf1 + TF);
  const char* rijB = (const char*)rij;
  const char* fijB = (const char*)fij;

  int t = waveId;
  if (t < nTiles) prefetch_tile(rijB, fijB, t, ldsR0, ldsF0, lane);
  int sel = 0;
  for (; t < nTiles; t += totalWaves) {
    const int  tn   = t + totalWaves;
    const bool more = (tn < nTiles);
    if (more)  // prefetch next tile into the other buffer
      prefetch_tile(rijB, fijB, tn, sel ? ldsR0 : ldsR1, sel ? ldsF0 : ldsF1, lane);
    // async loads complete in order: <=6 outstanding means current tile landed
    if (more) asm volatile("s_wait_asynccnt 6" ::: "memory");
    else      asm volatile("s_wait_asynccnt 0" ::: "memory");

    const float* bR = sel ? buf1 : buf0;
    const float* bF = bR + TF;
    const int ebase = t * TE;
#pragma unroll
    for (int k = 0; k < 4; ++k) {
      const int el = k * 32 + lane;            // lane-stride-12B LDS: conflict-free
      const int e  = ebase + el;
      const int s  = esrc[e];
      const int d  = edst[e];
      const int b  = batch[d];
      const float rx = bR[el * 3 + 0], ry = bR[el * 3 + 1], rz = bR[el * 3 + 2];
      const float fx = bF[el * 3 + 0], fy = bF[el * 3 + 1], fz = bF[el * 3 + 2];
      // force = segsum(fij, src) - segsum(fij, dst)
      fatomic(&force[s * 3 + 0],  fx);
      fatomic(&force[s * 3 + 1],  fy);
      fatomic(&force[s * 3 + 2],  fz);
      fatomic(&force[d * 3 + 0], -fx);
      fatomic(&force[d * 3 + 1], -fy);
      fatomic(&force[d * 3 + 2], -fz);
      // virial -> per-wave 32x6 bins (ds_add_f32); (6b+c) mod 64 injective in b
      atomicAdd(&myBins[b * 6 + 0], rx * fx);
      atomicAdd(&myBins[b * 6 + 1], ry * fy);
      atomicAdd(&myBins[b * 6 + 2], rz * fz);
      atomicAdd(&myBins[b * 6 + 3], rx * fy);
      atomicAdd(&myBins[b * 6 + 4], ry * fz);
      atomicAdd(&myBins[b * 6 + 5], rz * fx);
    }
    sel ^= 1;
  }

  __syncthreads();   // all waves done writing their private bins
  for (int i = tid; i < NBINS; i += BT) {
    float sum = 0.0f;
#pragma unroll
    for (int wv = 0; wv < WPB; ++wv) sum += sAcc[wv][i];
    fatomic(&acc[i], sum);
  }
}

__global__ void finalize_stress_kernel(const float* __restrict__ acc,
                                       const float* __restrict__ vol,
                                       float* __restrict__ stress) {
  const int i = threadIdx.x;
  if (i < NBINS) stress[i] = -acc[i] / vol[i / 6];
}

extern "C" void kernel_launch(void* const* d_in, const int* in_sizes, int n_in,
                              void* d_out, int out_size, void* d_ws, size_t ws_size,
                              hipStream_t stream) {
  (void)in_sizes; (void)n_in; (void)out_size; (void)ws_size;
  const float* rij   = (const float*)d_in[0];
  const float* fij   = (const float*)d_in[1];
  const float* vol   = (const float*)d_in[2];
  const int*   eidx  = (const int*)d_in[3];   // [2, E]
  const int*   batch = (const int*)d_in[4];

  float* out = (float*)d_out;                 // force [100000*3] ++ stress [32*6]
  float* acc = (float*)d_ws;                  // 192-float global accumulator

  hipMemsetAsync(out, 0, (size_t)K_NODES * 3 * sizeof(float), stream);
  hipMemsetAsync(acc, 0, (size_t)NBINS * sizeof(float), stream);

  edge_scatter_kernel<<<NB, BT, 0, stream>>>(rij, fij, eidx, eidx + K_EDGES,
                                             batch, out, acc);
  finalize_stress_kernel<<<1, NBINS, 0, stream>>>(acc, vol,
                                                  out + (size_t)K_NODES * 3);
}